// HybridVQC_14748917694978
// MI455X (gfx1250) — compile-verified
//
#include <hip/hip_runtime.h>
#include <hip/hip_bf16.h>
#include <math.h>

// ---------------------------------------------------------------------------
// Hybrid VQC: 8-qubit state-vector sim (one sample per wave32, state in VGPRs,
// cross-lane gates via shfl_xor) + final 3x3 linear layer done with
// V_WMMA_F32_16X16X4_F32 (exact f32 WMMA, K padded 3->4, N padded 3->16).
//
// Index convention: flattened amplitude index a in [0,256); reference wire i
// corresponds to bit p = 7 - i of a.  a = lane*8 + j:
//   bits 0..2 = j   (in-lane, register permutes)
//   bits 3..7 = lane (cross-lane, shfl_xor butterflies)
// ---------------------------------------------------------------------------

#define NQ 8
#define NL 3
#define BATCH 16384

typedef __attribute__((ext_vector_type(2))) float v2f;
typedef __attribute__((ext_vector_type(8))) float v8f;

__global__ __launch_bounds__(256) void vqc_circuit_kernel(
    const float* __restrict__ x,     // (BATCH, 8)
    const float* __restrict__ qw,    // (NL*2*NQ,) = 48
    float* __restrict__ zout)        // (BATCH, 3) staged in d_ws
{
    const int lane = threadIdx.x & 31;
    const int wave = threadIdx.x >> 5;
    const int s    = blockIdx.x * (blockDim.x >> 5) + wave;   // sample id

    // state: 8 complex amps per lane, a = lane*8 + j
    float re[8], im[8];
#pragma unroll
    for (int j = 0; j < 8; ++j) { re[j] = 0.f; im[j] = 0.f; }
    if (lane == 0) re[0] = 1.f;     // |0...0>

    const float PI = 3.14159265358979323846f;

#pragma unroll
    for (int layer = 0; layer < NL; ++layer) {
        const int base = layer * 2 * NQ;

        // ---------------- RY(x[i]*pi + qw[base+i]) on wire i ----------------
#pragma unroll
        for (int i = 0; i < NQ; ++i) {
            const float theta = x[s * NQ + i] * PI + qw[base + i];
            float sn, cs;
            __sincosf(0.5f * theta, &sn, &cs);
            const int p = 7 - i;
            if (p >= 3) {
                // lane-bit butterfly
                const int lmask = 1 << (p - 3);
                const float sgn = ((lane >> (p - 3)) & 1) ? sn : -sn;
#pragma unroll
                for (int j = 0; j < 8; ++j) {
                    const float orr = __shfl_xor(re[j], lmask, 32);
                    const float oii = __shfl_xor(im[j], lmask, 32);
                    // bit0 lane: c*mine - s*other ; bit1 lane: s*other + c*mine
                    re[j] = cs * re[j] + sgn * orr;
                    im[j] = cs * im[j] + sgn * oii;
                }
            } else {
                // in-register butterfly over j-bit p
                const int m = 1 << p;
#pragma unroll
                for (int j = 0; j < 8; ++j) {
                    if ((j & m) == 0) {
                        const int j1 = j | m;
                        const float r0 = re[j],  i0 = im[j];
                        const float r1 = re[j1], i1 = im[j1];
                        re[j]  = cs * r0 - sn * r1;  im[j]  = cs * i0 - sn * i1;
                        re[j1] = sn * r0 + cs * r1;  im[j1] = sn * i0 + cs * i1;
                    }
                }
            }
        }

        // ---------------- CNOT chain (0,1)..(6,7),(7,0) ----------------
        auto cnot = [&](int ctrl, int tgt) {
            const int pc = 7 - ctrl, pt = 7 - tgt;
            if (pc >= 3 && pt >= 3) {
                // both lane bits: exchange over tgt lane-bit, keep if ctrl bit==0
                const int lmask = 1 << (pt - 3);
                const int cbit  = (lane >> (pc - 3)) & 1;
#pragma unroll
                for (int j = 0; j < 8; ++j) {
                    const float orr = __shfl_xor(re[j], lmask, 32);
                    const float oii = __shfl_xor(im[j], lmask, 32);
                    if (cbit) { re[j] = orr; im[j] = oii; }
                }
            } else if (pc >= 3) {
                // ctrl on lane bit, tgt in-lane: local swap on ctrl==1 lanes
                const int cbit = (lane >> (pc - 3)) & 1;
                const int m = 1 << pt;
                if (cbit) {
#pragma unroll
                    for (int j = 0; j < 8; ++j) {
                        if ((j & m) == 0) {
                            const int j1 = j | m;
                            float t;
                            t = re[j]; re[j] = re[j1]; re[j1] = t;
                            t = im[j]; im[j] = im[j1]; im[j1] = t;
                        }
                    }
                }
            } else if (pt >= 3) {
                // ctrl in-lane, tgt on lane bit: exchange, keep where ctrl j-bit==0
                const int lmask = 1 << (pt - 3);
                const int m = 1 << pc;
#pragma unroll
                for (int j = 0; j < 8; ++j) {
                    const float orr = __shfl_xor(re[j], lmask, 32);
                    const float oii = __shfl_xor(im[j], lmask, 32);
                    if (j & m) { re[j] = orr; im[j] = oii; }
                }
            } else {
                // both in-lane
                const int mc = 1 << pc, mt = 1 << pt;
#pragma unroll
                for (int j = 0; j < 8; ++j) {
                    if ((j & mc) && !(j & mt)) {
                        const int j1 = j | mt;
                        float t;
                        t = re[j]; re[j] = re[j1]; re[j1] = t;
                        t = im[j]; im[j] = im[j1]; im[j1] = t;
                    }
                }
            }
        };
        cnot(0, 1); cnot(1, 2); cnot(2, 3); cnot(3, 4);
        cnot(4, 5); cnot(5, 6); cnot(6, 7); cnot(7, 0);

        // ---------------- RZ(qw[base+8+i]) on wire i ----------------
#pragma unroll
        for (int i = 0; i < NQ; ++i) {
            const float theta = qw[base + NQ + i];
            float s2, c2;
            __sincosf(0.5f * theta, &s2, &c2);
            const int p = 7 - i;
#pragma unroll
            for (int j = 0; j < 8; ++j) {
                const int a = lane * 8 + j;
                const float sgn = ((a >> p) & 1) ? s2 : -s2;  // e^{+-i theta/2}
                const float r = re[j], q = im[j];
                re[j] = r * c2 - q * sgn;
                im[j] = r * sgn + q * c2;
            }
        }
    }

    // ---------------- <Z_i> for wires 0..2 (lane bits 4,3,2) ----------------
    float pl = 0.f;
#pragma unroll
    for (int j = 0; j < 8; ++j) pl += re[j] * re[j] + im[j] * im[j];

    float z0, z1, z2;
#pragma unroll
    for (int i = 0; i < 3; ++i) {
        const int lb = 4 - i;                 // wire i -> bit 7-i -> lane bit 4-i
        float v = ((lane >> lb) & 1) ? -pl : pl;   // p(bit=0) - p(bit=1)
#pragma unroll
        for (int off = 16; off >= 1; off >>= 1) v += __shfl_xor(v, off, 32);
        if (i == 0) z0 = v; else if (i == 1) z1 = v; else z2 = v;
    }
    if (lane == 0) {
        zout[s * 3 + 0] = z0;
        zout[s * 3 + 1] = z1;
        zout[s * 3 + 2] = z2;
    }
}

// ---------------------------------------------------------------------------
// out[m, c] = sum_k z[m, k] * W[c, k] + b[c]  via V_WMMA_F32_16X16X4_F32.
// One wave handles 16 samples. ISA 32-bit layouts (7.12.2):
//   A 16x4: lanes 0-15 hold M=lane, VGPR{0,1}=K{0,1}; lanes 16-31: K{2,3}
//   B 4x16: lanes hold N=lane&15; lower lanes K{0,1}, upper lanes K{2,3}
//   C/D 16x16: VGPR r holds M=r (lanes 0-15) and M=r+8 (lanes 16-31), N=lane&15
// ---------------------------------------------------------------------------
__global__ __launch_bounds__(256) void linear_wmma_kernel(
    const float* __restrict__ z,     // (BATCH, 3)
    const float* __restrict__ W,     // (3, 3)
    const float* __restrict__ bias,  // (3,)
    float* __restrict__ out)         // (BATCH, 3)
{
    const int lane = threadIdx.x & 31;
    const int wave = threadIdx.x >> 5;
    const int tile = blockIdx.x * (blockDim.x >> 5) + wave;  // 16-sample tile
    const int half = lane >> 4;      // 0 -> K 0..1, 1 -> K 2..3
    const int nm   = lane & 15;      // A: row M, B/C/D: col N

    // A = z tile, K padded 3->4 with zeros
    const int m = tile * 16 + nm;
    v2f a;
    if (half == 0) { a.x = z[m * 3 + 0]; a.y = z[m * 3 + 1]; }
    else           { a.x = z[m * 3 + 2]; a.y = 0.f; }

    // B[k][n] = W[n][k] (W^T), zero-padded beyond 3x3
    v2f bmat;
    const int k0 = half * 2, k1 = half * 2 + 1;
    bmat.x = (nm < 3 && k0 < 3) ? W[nm * 3 + k0] : 0.f;
    bmat.y = (nm < 3 && k1 < 3) ? W[nm * 3 + k1] : 0.f;

    // C = bias broadcast down each column
    const float bb = (nm < 3) ? bias[nm] : 0.f;
    v8f c = {bb, bb, bb, bb, bb, bb, bb, bb};

    // D = A x B + C   (8 args: neg_a, A, neg_b, B, c_mod, C, reuse_a, reuse_b)
    v8f d = __builtin_amdgcn_wmma_f32_16x16x4_f32(
        false, a, false, bmat, (short)0, c, false, false);

    if (nm < 3) {
#pragma unroll
        for (int r = 0; r < 8; ++r) {
            const int row = tile * 16 + r + 8 * half;
            out[row * 3 + nm] = d[r];
        }
    }
}

extern "C" void kernel_launch(void* const* d_in, const int* in_sizes, int n_in,
                              void* d_out, int out_size, void* d_ws, size_t ws_size,
                              hipStream_t stream) {
    const float* x  = (const float*)d_in[0];   // (16384, 8)
    const float* qw = (const float*)d_in[1];   // (48,)
    const float* W  = (const float*)d_in[2];   // (3, 3)
    const float* b  = (const float*)d_in[3];   // (3,)
    float* out = (float*)d_out;                // (16384, 3)
    float* zbuf = (float*)d_ws;                // (16384, 3) scratch = 192 KB

    // 1 sample per wave, 8 waves (256 threads) per block
    vqc_circuit_kernel<<<BATCH / 8, 256, 0, stream>>>(x, qw, zbuf);

    // 16 samples per wave -> 1024 tiles, 8 waves per block
    linear_wmma_kernel<<<(BATCH / 16) / 8, 256, 0, stream>>>(zbuf, W, b, out);
}